// HGATConv_56788057588099
// MI455X (gfx1250) — compile-verified
//
#include <hip/hip_runtime.h>
#include <hip/hip_bf16.h>
#include <math.h>

// Problem constants (match reference)
constexpr int NN   = 50000;      // nodes
constexpr int EE   = 800000;     // edges (before self loops)
constexpr int ETOT = EE + NN;    // edges incl. self loops
constexpr int INF  = 128;        // input feature dim
constexpr int DD   = 64;         // output dim (1 time + 63 space)

typedef __attribute__((ext_vector_type(2))) float v2f;
typedef __attribute__((ext_vector_type(8))) float v8f;

// ---- monotone float <-> uint key for atomicMax over signed floats ----
__device__ __forceinline__ unsigned fkey(float f) {
    unsigned b = __float_as_uint(f);
    return (b & 0x80000000u) ? ~b : (b | 0x80000000u);
}
__device__ __forceinline__ float fdec(unsigned u) {
    unsigned b = (u & 0x80000000u) ? (u & 0x7FFFFFFFu) : ~u;
    return __uint_as_float(b);
}

// ---------------------------------------------------------------------
// K1: XS[N,64] = x[N,128] @ W[128,63] + b   via V_WMMA_F32_16X16X4_F32
//  - W staged to LDS with global_load_async_to_lds_b32, padded to 64 cols
//  - LDS layout: pair p = k/2 : lds[p*128 + col*2 + (k&1)]  -> one b64/frag
//  - 8 waves/block, each wave computes a 16-row x 64-col strip (4 tiles)
// ---------------------------------------------------------------------
__global__ __launch_bounds__(256) void k_gemm(const float* __restrict__ x,
                                              const float* __restrict__ W,
                                              const float* __restrict__ bias,
                                              float* __restrict__ XS) {
    __shared__ float ldsW[64 * 128];     // 8192 floats = 32 KB
    const int tid = threadIdx.x;

    // zero the pad entries (col 63 for every k)
    if (tid < 128) {
        const int k = tid;
        ldsW[(k >> 1) * 128 + 126 + (k & 1)] = 0.0f;
    }
    // async-stage W (128 x 63 floats, element e == byte e*4) global -> LDS
    for (int e = tid; e < 128 * 63; e += 256) {
        const int k   = e / 63;
        const int col = e - k * 63;
        const unsigned ldsOff =
            (unsigned)(uintptr_t)(ldsW + (k >> 1) * 128 + col * 2 + (k & 1));
        const unsigned gOff = (unsigned)(e * 4);
        asm volatile("global_load_async_to_lds_b32 %0, %1, %2"
                     :: "v"(ldsOff), "v"(gOff), "s"(W) : "memory");
    }
    asm volatile("s_wait_asynccnt 0x0" ::: "memory");
    __syncthreads();

    const int wave = tid >> 5;
    const int lane = tid & 31;
    const int tM   = blockIdx.x * 8 + wave;          // row tile (16 rows)
    if (tM < NN / 16) {                               // wave-uniform guard
        const int half = lane >> 4;                   // K-pair select
        const int l15  = lane & 15;
        const int row  = tM * 16 + l15;

        v8f acc[4];
        #pragma unroll
        for (int t = 0; t < 4; ++t) acc[t] = (v8f){};

        const v2f* xa = (const v2f*)(x + (size_t)row * INF);
        #pragma unroll 4
        for (int k = 0; k < INF; k += 4) {
            const int p = (k >> 1) + half;            // K pair index (ka = 2p)
            const v2f a     = xa[p];
            const v2f* brow = (const v2f*)(ldsW + p * 128);
            #pragma unroll
            for (int t = 0; t < 4; ++t) {
                const v2f b = brow[t * 16 + l15];
                acc[t] = __builtin_amdgcn_wmma_f32_16x16x4_f32(
                             false, a, false, b, (short)0, acc[t], false, false);
            }
        }
        #pragma unroll
        for (int t = 0; t < 4; ++t) {
            const int col = t * 16 + l15;
            if (col < 63) {                           // divergence only after WMMAs
                const float bb = bias[col];
                #pragma unroll
                for (int r = 0; r < 8; ++r) {
                    const int orow = tM * 16 + r + half * 8;   // C layout
                    XS[(size_t)orow * DD + col] = acc[t][r] + bb;
                }
            }
        }
    }
}

// ---------------------------------------------------------------------
// K2: per-node hyperbolic prep + init of all accumulators
// ---------------------------------------------------------------------
__global__ __launch_bounds__(256) void k_node(const float* __restrict__ XS,
                                              const float* __restrict__ att,
                                              float* __restrict__ OX,
                                              float* __restrict__ LOGX,
                                              float* __restrict__ AI,
                                              float* __restrict__ AJ,
                                              float* __restrict__ S1,
                                              float* __restrict__ S2,
                                              unsigned* __restrict__ M1,
                                              unsigned* __restrict__ M2,
                                              float* __restrict__ ACC) {
    const int n = blockIdx.x * blockDim.x + threadIdx.x;
    if (n >= NN) return;

    float sumsq = 0.0f;
    for (int d = 0; d < 63; ++d) {
        const float v = XS[n * DD + d];
        sumsq += v * v;
    }
    const float t     = sqrtf(sumsq + 1.0f);                 // C = 1
    const float nrm   = sqrtf(sumsq + 1e-15f);
    const float d0    = acoshf(fmaxf(t, 1.0f + 1e-6f));      // sqrt_c = 1
    const float scale = d0 / nrm;

    OX[n * DD + 0]   = t;
    LOGX[n * DD + 0] = 0.0f;
    float ai = 0.0f, aj = 0.0f;                              // logx[0]==0
    for (int d = 0; d < 63; ++d) {
        const float xs = XS[n * DD + d];
        const float lx = scale * xs;
        OX[n * DD + 1 + d]   = xs;
        LOGX[n * DD + 1 + d] = lx;
        ai += lx * att[1 + d];
        aj += lx * att[DD + 1 + d];
    }
    AI[n] = ai; AJ[n] = aj;
    S1[n] = 0.0f; S2[n] = 0.0f;
    M1[n] = 0u;   M2[n] = 0u;                                // below fkey(-inf)
    for (int d = 0; d < DD; ++d) ACC[n * DD + d] = 0.0f;     // d_out accumulator
}

// ---------------------------------------------------------------------
// K3: per-edge Minkowski inner product (wave per edge), sqd, raw alpha,
//     segment max #1 (keyed by i)
// ---------------------------------------------------------------------
__global__ __launch_bounds__(256) void k_edge1(const int* __restrict__ ei,
                                               const float* __restrict__ OX,
                                               const float* __restrict__ AI,
                                               const float* __restrict__ AJ,
                                               float* __restrict__ SQD,
                                               float* __restrict__ ALPHA,
                                               unsigned* __restrict__ M1) {
    const int wave = (blockIdx.x * blockDim.x + threadIdx.x) >> 5;
    const int lane = threadIdx.x & 31;
    if (wave >= ETOT) return;

    int i, j;
    if (wave < EE) { i = ei[wave]; j = ei[EE + wave]; }
    else           { i = wave - EE; j = i; }

    const float2 a = ((const float2*)(OX + (size_t)i * DD))[lane];
    const float2 b = ((const float2*)(OX + (size_t)j * DD))[lane];
    float p = a.x * b.x + a.y * b.y;
    if (lane == 0) p -= 2.0f * a.x * b.x;        // time component gets minus sign
    #pragma unroll
    for (int off = 16; off > 0; off >>= 1) p += __shfl_xor(p, off, 32);

    if (lane == 0) {
        const float z   = fmaxf(-p, 1.0f + 1e-6f);   // -inner / C, C = 1
        const float ach = acoshf(z);
        const float sqd = ach * ach;
        SQD[wave]   = sqd;
        ALPHA[wave] = AI[i] + AJ[j];
        atomicMax(&M1[i], fkey(sqd));
    }
}

// ---------------------------------------------------------------------
// K4/K6: e = exp(v - max[i]); sum into S[i]; overwrite V with e
// ---------------------------------------------------------------------
__global__ __launch_bounds__(256) void k_expsum(const int* __restrict__ ei,
                                                float* __restrict__ V,
                                                const unsigned* __restrict__ M,
                                                float* __restrict__ S) {
    const int e = blockIdx.x * blockDim.x + threadIdx.x;
    if (e >= ETOT) return;
    const int i = (e < EE) ? ei[e] : (e - EE);
    const float ex = __expf(V[e] - fdec(M[i]));
    V[e] = ex;
    atomicAdd(&S[i], ex);
}

// ---------------------------------------------------------------------
// K5: alpha = leaky( alpha_raw * softmax1 ); segment max #2
// ---------------------------------------------------------------------
__global__ __launch_bounds__(256) void k_edge3(const int* __restrict__ ei,
                                               const float* __restrict__ SQD,
                                               const float* __restrict__ S1,
                                               float* __restrict__ ALPHA,
                                               unsigned* __restrict__ M2) {
    const int e = blockIdx.x * blockDim.x + threadIdx.x;
    if (e >= ETOT) return;
    const int i = (e < EE) ? ei[e] : (e - EE);
    const float w1 = SQD[e] / (S1[i] + 1e-16f);
    float a = ALPHA[e] * w1;
    a = (a >= 0.0f) ? a : 0.2f * a;              // leaky relu, NEG = 0.2
    ALPHA[e] = a;
    atomicMax(&M2[i], fkey(a));
}

// ---------------------------------------------------------------------
// K7: out[i] += softmax2(e) * logx[j]  (64 lanes per edge, coalesced)
// ---------------------------------------------------------------------
__global__ __launch_bounds__(256) void k_scatter(const int* __restrict__ ei,
                                                 const float* __restrict__ ALPHA,
                                                 const float* __restrict__ S2,
                                                 const float* __restrict__ LOGX,
                                                 float* __restrict__ ACC) {
    const int gid = blockIdx.x * blockDim.x + threadIdx.x;
    if (gid >= ETOT * DD) return;
    const int e = gid >> 6;
    const int d = gid & 63;
    int i, j;
    if (e < EE) { i = ei[e]; j = ei[EE + e]; }
    else        { i = e - EE; j = i; }
    const float w2 = ALPHA[e] / (S2[i] + 1e-16f);
    atomicAdd(&ACC[(size_t)i * DD + d], w2 * LOGX[(size_t)j * DD + d]);
}

// ---------------------------------------------------------------------
// K8: per-node finalize: relu, exp-map (sinh), time component
// ---------------------------------------------------------------------
__global__ __launch_bounds__(256) void k_final(float* __restrict__ OUT) {
    const int n = blockIdx.x * blockDim.x + threadIdx.x;
    if (n >= NN) return;
    float ss = 0.0f;
    for (int d = 1; d < DD; ++d) {
        const float v = fmaxf(OUT[n * DD + d], 0.0f);
        ss += v * v;
    }
    const float un = sqrtf(ss + 1e-15f);
    const float s  = sinhf(un) / un;             // sqrt_c = 1
    float tss = 0.0f;
    for (int d = 1; d < DD; ++d) {
        const float v  = fmaxf(OUT[n * DD + d], 0.0f);
        const float sp = s * v;
        OUT[n * DD + d] = sp;
        tss += sp * sp;
    }
    OUT[n * DD + 0] = sqrtf(tss + 1.0f);         // time = sqrt(|sp|^2 + C)
}

// ---------------------------------------------------------------------
extern "C" void kernel_launch(void* const* d_in, const int* in_sizes, int n_in,
                              void* d_out, int out_size, void* d_ws, size_t ws_size,
                              hipStream_t stream) {
    const float* x    = (const float*)d_in[0];   // (N, 128)
    const float* W    = (const float*)d_in[1];   // (128, 63)
    const float* bias = (const float*)d_in[2];   // (63,)
    const float* att  = (const float*)d_in[3];   // (1, 1, 128)
    const int*   ei   = (const int*)d_in[4];     // (2, E) row-major
    float*       out  = (float*)d_out;           // (N, 64); also accumulator

    // workspace layout (floats)
    float* p = (float*)d_ws;
    float* XS   = p; p += (size_t)NN * DD;       // GEMM result (63 cols used)
    float* OX   = p; p += (size_t)NN * DD;       // [t, xs]
    float* LOGX = p; p += (size_t)NN * DD;       // [0, d0*xs/nrm]
    float* AI   = p; p += NN;
    float* AJ   = p; p += NN;
    float* S1   = p; p += NN;
    float* S2   = p; p += NN;
    float* SQD  = p; p += ETOT;                  // sqd -> exp1
    float* ALPH = p; p += ETOT;                  // raw alpha -> leaky -> exp2
    unsigned* M1 = (unsigned*)p; p += NN;
    unsigned* M2 = (unsigned*)p; p += NN;

    const int TB = 256;

    k_gemm<<<dim3((NN / 16 + 7) / 8), dim3(TB), 0, stream>>>(x, W, bias, XS);
    k_node<<<dim3((NN + TB - 1) / TB), dim3(TB), 0, stream>>>(XS, att, OX, LOGX,
                                                              AI, AJ, S1, S2, M1, M2, out);
    k_edge1<<<dim3((ETOT * 32 + TB - 1) / TB), dim3(TB), 0, stream>>>(ei, OX, AI, AJ,
                                                                      SQD, ALPH, M1);
    k_expsum<<<dim3((ETOT + TB - 1) / TB), dim3(TB), 0, stream>>>(ei, SQD, M1, S1);
    k_edge3<<<dim3((ETOT + TB - 1) / TB), dim3(TB), 0, stream>>>(ei, SQD, S1, ALPH, M2);
    k_expsum<<<dim3((ETOT + TB - 1) / TB), dim3(TB), 0, stream>>>(ei, ALPH, M2, S2);
    k_scatter<<<dim3((ETOT * DD + TB - 1) / TB), dim3(TB), 0, stream>>>(ei, ALPH, S2,
                                                                        LOGX, out);
    k_final<<<dim3((NN + TB - 1) / TB), dim3(TB), 0, stream>>>(out);
}